// GraphSAGE_51084341019062
// MI455X (gfx1250) — compile-verified
//
#include <hip/hip_runtime.h>
#include <hip/hip_bf16.h>

typedef float v2f __attribute__((ext_vector_type(2)));
typedef float v8f __attribute__((ext_vector_type(8)));

#define FDIM 128
#define KSTEPS 32           // 128 / 4 (K per WMMA f32 op)

// ---------------- zero scratch ----------------
__global__ void sage_zero_kernel(float* __restrict__ p, long long count4) {
    // count4 = number of floats, guaranteed multiple of 4
    long long i = ((long long)blockIdx.x * blockDim.x + threadIdx.x) * 4;
    long long stride = (long long)gridDim.x * blockDim.x * 4;
    float4 z = make_float4(0.f, 0.f, 0.f, 0.f);
    for (; i < count4; i += stride) {
        *(float4*)(p + i) = z;
    }
}

// ---------------- in-degree (float, shared by both layers) ----------------
__global__ void sage_degree_kernel(const int* __restrict__ dst,
                                   float* __restrict__ deg, long long E) {
    long long i = (long long)blockIdx.x * blockDim.x + threadIdx.x;
    if (i < E) {
        atomicAdd(&deg[dst[i]], 1.0f);
    }
}

// ---------------- edge scatter: hsum[dst] += feat[src] ----------------
// one wave (32 lanes) per edge; each lane moves 4 consecutive floats
__global__ void sage_scatter_kernel(const float* __restrict__ feat,
                                    const int* __restrict__ src,
                                    const int* __restrict__ dst,
                                    float* __restrict__ hsum, long long E) {
    int lane = threadIdx.x & 31;
    int wv   = threadIdx.x >> 5;
    long long e = (long long)blockIdx.x * (blockDim.x >> 5) + wv;
    if (e >= E) return;
    int s = src[e];
    int d = dst[e];
    const float4 v = *(const float4*)(feat + (long long)s * FDIM + lane * 4);
    float* o = hsum + (long long)d * FDIM + lane * 4;
    atomicAdd(o + 0, v.x);
    atomicAdd(o + 1, v.y);
    atomicAdd(o + 2, v.z);
    atomicAdd(o + 3, v.w);
}

// ---------------- mean normalization: hsum[r,:] *= 1/max(deg[r],1) ----------------
// done once per layer so the WMMA loop stays pure load+wmma (no 8x-duplicated
// per-column-tile rescaling, no divides in the GEMM hot loop)
__global__ void sage_normalize_kernel(float* __restrict__ hsum,
                                      const float* __restrict__ deg,
                                      long long n_nodes) {
    long long i = (long long)blockIdx.x * blockDim.x + threadIdx.x;  // (row, 16B chunk)
    long long total = n_nodes * (FDIM / 4);
    if (i >= total) return;
    long long row = i >> 5;                 // 32 float4 chunks per row
    float rd = 1.0f / fmaxf(deg[row], 1.0f);
    float4* p = (float4*)(hsum) + i;
    float4 v = *p;
    v.x *= rd; v.y *= rd; v.z *= rd; v.w *= rd;
    *p = v;
}

// ---------------- fused SAGEConv GEMM via V_WMMA_F32_16X16X4_F32 ----------------
// out[r, :] = A[r, :] @ Ws + HN[r, :] @ Wn + bias   (opt. relu)
// block = 256 threads = 8 waves; wave w owns output columns [16w, 16w+16)
// B fragments for both weight matrices are register-resident across the row loop.
__global__ __launch_bounds__(256) void sage_gemm_kernel(
    const float* __restrict__ A,    // [N,128] self features
    const float* __restrict__ HN,   // [N,128] normalized neighbor means
    const float* __restrict__ Ws,   // [128,128] row-major
    const float* __restrict__ Wn,   // [128,128] row-major
    const float* __restrict__ bias, // [128]
    float* __restrict__ out,        // [N,128]
    int n_row_tiles, int relu) {
    const int lane = threadIdx.x & 31;
    const int wv   = threadIdx.x >> 5;   // 0..7 -> column tile
    const int n0   = wv * 16;
    const int m    = lane & 15;          // N index within tile (B/C), M row (A)
    const int half = lane >> 4;          // K sub-select

    // ---- preload B fragments (K={2h,2h+1} in .x/.y per VGPR pair, per ISA layout) ----
    v2f bs[KSTEPS], bn[KSTEPS];
#pragma unroll
    for (int kk = 0; kk < KSTEPS; ++kk) {
        const int k0  = kk * 4 + 2 * half;
        const int col = n0 + m;
        bs[kk].x = Ws[(k0 + 0) * FDIM + col];
        bs[kk].y = Ws[(k0 + 1) * FDIM + col];
        bn[kk].x = Wn[(k0 + 0) * FDIM + col];
        bn[kk].y = Wn[(k0 + 1) * FDIM + col];
    }
    const float bb = bias[n0 + m];

    for (int t = blockIdx.x; t < n_row_tiles; t += gridDim.x) {
        const int r0 = t * 16;
        const float* arow = A  + (size_t)(r0 + m) * FDIM + 2 * half;
        const float* hrow = HN + (size_t)(r0 + m) * FDIM + 2 * half;

        v8f c = {};
#pragma unroll
        for (int kk = 0; kk < KSTEPS; ++kk) {
            v2f ax = *(const v2f*)(arow + kk * 4);   // A frag: K={2h,2h+1}+4kk
            v2f ah = *(const v2f*)(hrow + kk * 4);
            c = __builtin_amdgcn_wmma_f32_16x16x4_f32(
                    false, ax, false, bs[kk], (short)0, c, false, false);
            c = __builtin_amdgcn_wmma_f32_16x16x4_f32(
                    false, ah, false, bn[kk], (short)0, c, false, false);
        }

        // C layout: VGPR j -> row r0 + j + 8*half, col n0 + m
#pragma unroll
        for (int j = 0; j < 8; ++j) {
            float v = c[j] + bb;
            if (relu) v = fmaxf(v, 0.0f);
            out[(size_t)(r0 + j + 8 * half) * FDIM + (n0 + m)] = v;
        }
    }
}

extern "C" void kernel_launch(void* const* d_in, const int* in_sizes, int n_in,
                              void* d_out, int out_size, void* d_ws, size_t ws_size,
                              hipStream_t stream) {
    const float* x   = (const float*)d_in[0];
    const int*   src = (const int*)d_in[1];
    const int*   dst = (const int*)d_in[2];
    const float* W1s = (const float*)d_in[3];
    const float* W1n = (const float*)d_in[4];
    const float* b1  = (const float*)d_in[5];
    const float* W2s = (const float*)d_in[6];
    const float* W2n = (const float*)d_in[7];
    const float* b2  = (const float*)d_in[8];
    float* out = (float*)d_out;

    const long long E = in_sizes[1];
    const long long N = (long long)in_sizes[0] / FDIM;   // 100000
    const int n_row_tiles = (int)(N / 16);               // 6250 exact

    // workspace layout (floats): deg[N] | hsum[N*128] | h1[N*128]
    const size_t HSUM_OFF = 131072;                      // 512KB-aligned region for deg
    float* ws   = (float*)d_ws;
    float* deg  = ws;
    float* hsum = ws + HSUM_OFF;
    float* h1   = ws + HSUM_OFF + (size_t)N * FDIM;

    const int ZB = 2048, THR = 256;
    const long long zero_all = (long long)HSUM_OFF + N * FDIM;  // deg + hsum
    const int norm_blocks = (int)((N * (FDIM / 4) + THR - 1) / THR);

    // ---- layer 1 ----
    sage_zero_kernel<<<ZB, THR, 0, stream>>>(ws, zero_all);
    sage_degree_kernel<<<(int)((E + THR - 1) / THR), THR, 0, stream>>>(dst, deg, E);
    sage_scatter_kernel<<<(int)((E + 7) / 8), THR, 0, stream>>>(x, src, dst, hsum, E);
    sage_normalize_kernel<<<norm_blocks, THR, 0, stream>>>(hsum, deg, N);
    sage_gemm_kernel<<<512, THR, 0, stream>>>(x, hsum, W1s, W1n, b1, h1,
                                              n_row_tiles, 1);

    // ---- layer 2 ----
    sage_zero_kernel<<<ZB, THR, 0, stream>>>(hsum, N * FDIM);
    sage_scatter_kernel<<<(int)((E + 7) / 8), THR, 0, stream>>>(h1, src, dst, hsum, E);
    sage_normalize_kernel<<<norm_blocks, THR, 0, stream>>>(hsum, deg, N);
    sage_gemm_kernel<<<512, THR, 0, stream>>>(h1, hsum, W2s, W2n, b2, out,
                                              n_row_tiles, 0);
}